// DTWLayer_33406255628865
// MI455X (gfx1250) — compile-verified
//
#include <hip/hip_runtime.h>

// MI455X (gfx1250, wave32) subsequence-DTW.
// Kernel 1 (per-batch workgroup, 32 waves):
//   - per 16-row chunk: fp32 WMMA (16x16x4) computes C tile into 256KB LDS,
//     then 16 min-plus scan rows (block cumsum + cummin via wave shuffles),
//     emitting 2-bit predecessor codes to global (byte per 4 cells).
//   - final row: block min/argmin -> dtw_cost + start column.
// Kernel 2: serial backtrack over predecessor codes (1 lane per batch).
// Kernel 3: gather w_ts / w_vs.
//
// v2: batch the 16 B-operand loads per tile into one clause before the WMMA
//     chain (previously the compiler serialized load->wait->wmma x16), and
//     hoist the 8 x2 row-norm LDS reads to chunk scope.

typedef float v2f __attribute__((ext_vector_type(2)));
typedef float v8f __attribute__((ext_vector_type(8)));

#define FINF __builtin_huge_valf()

constexpr int B = 16, N = 1024, M = 4096, Dd = 64;
constexpr int THREADS = 1024;            // 32 waves per workgroup
constexpr int WAVES = THREADS / 32;      // 32
constexpr int CHUNK = 16;                // rows per WMMA/DP chunk
constexpr int NCHUNK = N / CHUNK;        // 64
constexpr int JPT = M / THREADS;         // 4 columns owned per thread

__device__ __forceinline__ float wave_scan_add(float v, int lane) {
#pragma unroll
  for (int d = 1; d < 32; d <<= 1) {
    float n = __shfl_up(v, d, 32);
    if (lane >= d) v += n;
  }
  return v;
}
__device__ __forceinline__ float wave_scan_min(float v, int lane) {
#pragma unroll
  for (int d = 1; d < 32; d <<= 1) {
    float n = __shfl_up(v, d, 32);
    if (lane >= d) v = fminf(v, n);
  }
  return v;
}

__global__ __launch_bounds__(THREADS) void dtw_fwd_kernel(
    const float* __restrict__ x, const float* __restrict__ y,
    float* __restrict__ cost_out, unsigned char* __restrict__ ptrBuf,
    int* __restrict__ j0Buf) {
  extern __shared__ float smem[];
  float* Ctile = smem;                    // CHUNK * M          (256 KB)
  float* Dbuf  = Ctile + CHUNK * M;       // 2 * M  (prev/cur D rows)
  float* y2s   = Dbuf + 2 * M;            // M
  float* x2s   = y2s + M;                 // N
  float* scanS = x2s + N;                 // WAVES
  float* scanM = scanS + WAVES;           // WAVES
  float* redV  = scanM + WAVES;           // WAVES
  int*   redI  = (int*)(redV + WAVES);    // WAVES

  const int b    = blockIdx.x;
  const int tid  = threadIdx.x;
  const int lane = tid & 31;
  const int wid  = tid >> 5;
  const int half = lane >> 4;             // WMMA lane halves (K split)
  const int l16  = lane & 15;

  const float* xb = x + (size_t)b * N * Dd;
  const float* yb = y + (size_t)b * M * Dd;
  unsigned char* pb = ptrBuf + (size_t)b * N * (M / 4);

  // ---- norms ----
  {
    const float4* xr = (const float4*)(xb + tid * Dd);
    float s = 0.f;
#pragma unroll
    for (int k = 0; k < Dd / 4; ++k) {
      float4 v = xr[k];
      s += v.x * v.x + v.y * v.y + v.z * v.z + v.w * v.w;
    }
    x2s[tid] = s;
  }
  for (int j = tid; j < M; j += THREADS) {
    const float4* yr = (const float4*)(yb + j * Dd);
    float s = 0.f;
#pragma unroll
    for (int k = 0; k < Dd / 4; ++k) {
      float4 v = yr[k];
      s += v.x * v.x + v.y * v.y + v.z * v.z + v.w * v.w;
    }
    y2s[j] = s;
  }
  __syncthreads();

  float lastD[JPT];
#pragma unroll 1
  for (int ci = 0; ci < NCHUNK; ++ci) {
    // ---------- Phase 1: C tile via fp32 WMMA ----------
    // A (16x4 f32) layout: lane l16 = row; VGPR0/1 = K = 2*half + {0,1}.
    v2f A[16];
    float x2row[8];
    {
      const float* xrow = xb + (size_t)(ci * CHUNK + l16) * Dd;
#pragma unroll
      for (int kk = 0; kk < 16; ++kk)
        A[kk] = *(const v2f*)(xrow + 4 * kk + 2 * half);
#pragma unroll
      for (int r = 0; r < 8; ++r)
        x2row[r] = x2s[ci * CHUNK + r + 8 * half];
    }
#pragma unroll 1
    for (int tIdx = 0; tIdx < 8; ++tIdx) {
      const int mt  = wid + WAVES * tIdx;        // column tile 0..255
      const int col = mt * 16 + l16;
      const float* ycol = yb + (size_t)col * Dd; // B[k][n] = y[col][k]

      // batch all 16 B loads (one clause, one wait), then 16 WMMAs
      v2f Bv[16];
#pragma unroll
      for (int kk = 0; kk < 16; ++kk)
        Bv[kk] = *(const v2f*)(ycol + 4 * kk + 2 * half);

      v8f acc = {0.f, 0.f, 0.f, 0.f, 0.f, 0.f, 0.f, 0.f};
#pragma unroll
      for (int kk = 0; kk < 16; ++kk)
        acc = __builtin_amdgcn_wmma_f32_16x16x4_f32(
            false, A[kk], false, Bv[kk], (short)0, acc, false, false);

      const float y2v = y2s[col];
#pragma unroll
      for (int r = 0; r < 8; ++r) {
        const int rl = r + 8 * half;             // C/D row layout
        Ctile[rl * M + col] = x2row[r] + y2v - 2.0f * acc[r];
      }
    }
    __syncthreads();

    // ---------- Phase 2: 16 DP rows (min-plus scan) ----------
#pragma unroll 1
    for (int rl = 0; rl < CHUNK; ++rl) {
      const int i  = ci * CHUNK + rl;
      const int j0 = tid * JPT;
      float4 c4 = *(const float4*)(Ctile + rl * M + j0);
      float c[JPT] = {c4.x, c4.y, c4.z, c4.w};

      float* Dprev = Dbuf + (((i + 1) & 1) * M);
      float* Dcur  = Dbuf + ((i & 1) * M);

      if (i == 0) {
        *(float4*)(Dcur + j0) = c4;
#pragma unroll
        for (int k = 0; k < JPT; ++k) lastD[k] = c[k];
        __syncthreads();
        continue;
      }

      float dp[JPT];
#pragma unroll
      for (int k = 0; k < JPT; ++k) dp[k] = Dprev[j0 + k];
      const float dpm1 = (j0 == 0) ? FINF : Dprev[j0 - 1];

      // a = min(Dprev, shift(Dprev)) + c ; S = cumsum(c) ; T = cummin(a - S)
      float a[JPT], S[JPT], Tm[JPT];
      float run = 0.f;
#pragma unroll
      for (int k = 0; k < JPT; ++k) {
        const float dshift = (k == 0) ? dpm1 : dp[k - 1];
        a[k] = fminf(dp[k], dshift) + c[k];
        run += c[k];
        S[k] = run;
      }
      float tmin = FINF;
#pragma unroll
      for (int k = 0; k < JPT; ++k) {
        tmin = fminf(tmin, a[k] - S[k]);
        Tm[k] = tmin;
      }
      // block scans: wave shuffles + one 32-entry LDS hop (2 barriers)
      const float incS = wave_scan_add(run, lane);
      const float incM = wave_scan_min(tmin, lane);
      if (lane == 31) { scanS[wid] = incS; scanM[wid] = incM; }
      __syncthreads();
      if (wid == 0) {
        scanS[lane] = wave_scan_add(scanS[lane], lane);
        scanM[lane] = wave_scan_min(scanM[lane], lane);
      }
      __syncthreads();
      float exS = __shfl_up(incS, 1, 32); if (lane == 0) exS = 0.f;
      float exM = __shfl_up(incM, 1, 32); if (lane == 0) exM = FINF;
      const float baseS = (wid > 0 ? scanS[wid - 1] : 0.f) + exS;
      const float baseM = fminf((wid > 0 ? scanM[wid - 1] : FINF), exM);

      float Dv[JPT];
#pragma unroll
      for (int k = 0; k < JPT; ++k)
        Dv[k] = (baseS + S[k]) + fminf(baseM, Tm[k]);
      *(float4*)(Dcur + j0) = make_float4(Dv[0], Dv[1], Dv[2], Dv[3]);
      __syncthreads();

      // predecessor codes (reference tie order: diag > up > left)
      const float dlm1 = (j0 == 0) ? FINF : Dcur[j0 - 1];
      unsigned int code = 0;
#pragma unroll
      for (int k = 0; k < JPT; ++k) {
        const float dd = (k == 0) ? dpm1 : dp[k - 1];
        const float du = dp[k];
        const float dl = (k == 0) ? dlm1 : Dv[k - 1];
        const float best = fminf(fminf(dd, du), dl);
        const unsigned int ck = (dd == best) ? 0u : ((du == best) ? 1u : 2u);
        code |= ck << (2 * k);
      }
      pb[(size_t)i * (M / 4) + tid] = (unsigned char)code;
      if (i == N - 1) {
#pragma unroll
        for (int k = 0; k < JPT; ++k) lastD[k] = Dv[k];
      }
    }
  }

  // ---- min/argmin over final row (first occurrence on ties) ----
  float v = lastD[0];
  int idx = tid * JPT;
#pragma unroll
  for (int k = 1; k < JPT; ++k)
    if (lastD[k] < v) { v = lastD[k]; idx = tid * JPT + k; }
#pragma unroll
  for (int d = 16; d >= 1; d >>= 1) {
    float ov = __shfl_down(v, d, 32);
    int   oi = __shfl_down(idx, d, 32);
    if (ov < v || (ov == v && oi < idx)) { v = ov; idx = oi; }
  }
  if (lane == 0) { redV[wid] = v; redI[wid] = idx; }
  __syncthreads();
  if (wid == 0) {
    float vv = redV[lane];
    int ii = redI[lane];
#pragma unroll
    for (int d = 16; d >= 1; d >>= 1) {
      float ov = __shfl_down(vv, d, 32);
      int   oi = __shfl_down(ii, d, 32);
      if (ov < vv || (ov == vv && oi < ii)) { vv = ov; ii = oi; }
    }
    if (lane == 0) { cost_out[b] = vv; j0Buf[b] = ii; }
  }
}

__global__ void dtw_backtrack_kernel(const unsigned char* __restrict__ ptrBuf,
                                     const int* __restrict__ j0Buf,
                                     int* __restrict__ ylBuf) {
  const int b = blockIdx.x;
  const int lane = threadIdx.x;
  int* yl = ylBuf + b * N;
  for (int t = lane; t < N; t += 32) yl[t] = 0;
  __syncthreads();
  if (lane != 0) return;
  const unsigned char* pb = ptrBuf + (size_t)b * N * (M / 4);
  int i = N - 1;
  int j = j0Buf[b];
  yl[i] = j;  // first arrival at a row carries the largest j (j non-increasing)
  while (i > 0) {
    const unsigned int byte = pb[(size_t)i * (M / 4) + (j >> 2)];
    const unsigned int ck = (byte >> ((j & 3) * 2)) & 3u;
    if (ck == 0u)      { --i; --j; yl[i] = j; }   // diag
    else if (ck == 1u) { --i;      yl[i] = j; }   // up
    else               { --j; }                   // left (same row, keep max)
  }
}

__global__ void dtw_output_kernel(const float* __restrict__ x_t,
                                  const float* __restrict__ y_t,
                                  const int* __restrict__ ylBuf,
                                  float* __restrict__ w_ts,
                                  float* __restrict__ w_vs) {
  const int idx = blockIdx.x * blockDim.x + threadIdx.x;
  if (idx >= B * N) return;
  const int b = idx / N;
  w_ts[idx] = x_t[idx];
  w_vs[idx] = y_t[b * M + ylBuf[idx]];
}

extern "C" void kernel_launch(void* const* d_in, const int* in_sizes, int n_in,
                              void* d_out, int out_size, void* d_ws, size_t ws_size,
                              hipStream_t stream) {
  const float* x   = (const float*)d_in[0];
  const float* y   = (const float*)d_in[1];
  const float* x_t = (const float*)d_in[2];
  const float* y_t = (const float*)d_in[3];

  float* out  = (float*)d_out;
  float* cost = out;            // [B]
  float* w_ts = out + B;        // [B*N]
  float* w_vs = w_ts + B * N;   // [B*N]

  unsigned char* ptrBuf = (unsigned char*)d_ws;                 // 16 MiB
  int* j0Buf = (int*)(ptrBuf + (size_t)B * N * (M / 4));        // B ints
  int* ylBuf = j0Buf + B;                                        // B*N ints

  const size_t lds_bytes =
      (size_t)(CHUNK * M + 2 * M + M + N + 4 * WAVES) * sizeof(float);  // ~309 KB

  dtw_fwd_kernel<<<B, THREADS, lds_bytes, stream>>>(x, y, cost, ptrBuf, j0Buf);
  dtw_backtrack_kernel<<<B, 32, 0, stream>>>(ptrBuf, j0Buf, ylBuf);
  const int tot = B * N;
  dtw_output_kernel<<<(tot + 255) / 256, 256, 0, stream>>>(x_t, y_t, ylBuf,
                                                           w_ts, w_vs);
}